// DynamicToolEmbedding_70489003261953
// MI455X (gfx1250) — compile-verified
//
#include <hip/hip_runtime.h>

#define VOCAB_SZ  32000
#define NUM_NEW   512
#define HIDDEN    4096
#define PROF_DIM  64
#define MLP_HID   256
#define BATCH     4
#define SEQ       4096

typedef __attribute__((ext_vector_type(2))) float v2f;
typedef __attribute__((ext_vector_type(4))) float v4f;
typedef __attribute__((ext_vector_type(8))) float v8f;

// D = A(16x4 f32) * B(4x16 f32) + C  --  v_wmma_f32_16x16x4_f32
__device__ __forceinline__ v8f wmma_k4(v2f a, v2f b, v8f c) {
    return __builtin_amdgcn_wmma_f32_16x16x4_f32(
        /*neg_a=*/false, a, /*neg_b=*/false, b,
        /*c_mod=*/(short)0, c, /*reuse_a=*/false, /*reuse_b=*/false);
}

// ---------------------------------------------------------------------------
// Stage 1: H[512 x 256] = relu(profiles[512 x 64] @ W1[64 x 256] + b1)
// One wave32 per 16x16 output tile.
// ---------------------------------------------------------------------------
__global__ __launch_bounds__(32) void mlp1_kernel(
    const float* __restrict__ prof, const float* __restrict__ W1,
    const float* __restrict__ b1, float* __restrict__ H) {
    const int tm   = blockIdx.x * 16;
    const int tn   = blockIdx.y * 16;
    const int lane = threadIdx.x & 31;
    const int half = lane >> 4;     // 0: lanes 0-15, 1: lanes 16-31
    const int l    = lane & 15;

    v8f acc = {};
#pragma unroll
    for (int k = 0; k < PROF_DIM; k += 4) {
        const int kk = k + 2 * half;
        v2f a, b;
        a.x = prof[(tm + l) * PROF_DIM + kk];       // A[m][kk]
        a.y = prof[(tm + l) * PROF_DIM + kk + 1];   // A[m][kk+1]
        b.x = W1[(kk)     * MLP_HID + tn + l];      // B[kk][n]
        b.y = W1[(kk + 1) * MLP_HID + tn + l];      // B[kk+1][n]
        acc = wmma_k4(a, b, acc);
    }

    const float bias = b1[tn + l];
#pragma unroll
    for (int r = 0; r < 8; ++r) {
        const int m = tm + r + 8 * half;            // C/D row mapping
        float v = acc[r] + bias;
        H[m * MLP_HID + tn + l] = v > 0.0f ? v : 0.0f;
    }
}

// ---------------------------------------------------------------------------
// Stage 2: combined[512 x 4096] = H[512 x 256] @ W2[256 x 4096] + b2 + sem
// One wave32 per 16x16 output tile, 64 WMMA steps over K=256.
// ---------------------------------------------------------------------------
__global__ __launch_bounds__(32) void mlp2_kernel(
    const float* __restrict__ H, const float* __restrict__ W2,
    const float* __restrict__ b2, const float* __restrict__ sem,
    float* __restrict__ combined) {
    const int tm   = blockIdx.x * 16;
    const int tn   = blockIdx.y * 16;
    const int lane = threadIdx.x & 31;
    const int half = lane >> 4;
    const int l    = lane & 15;

    v8f acc = {};
#pragma unroll 8
    for (int k = 0; k < MLP_HID; k += 4) {
        const int kk = k + 2 * half;
        v2f a, b;
        a.x = H[(tm + l) * MLP_HID + kk];
        a.y = H[(tm + l) * MLP_HID + kk + 1];
        b.x = W2[(size_t)(kk)     * HIDDEN + tn + l];
        b.y = W2[(size_t)(kk + 1) * HIDDEN + tn + l];
        acc = wmma_k4(a, b, acc);
    }

    const float bias = b2[tn + l];
#pragma unroll
    for (int r = 0; r < 8; ++r) {
        const int m = tm + r + 8 * half;
        const size_t idx = (size_t)m * HIDDEN + tn + l;
        combined[idx] = acc[r] + bias + sem[idx];
    }
}

// ---------------------------------------------------------------------------
// Stage 3: gather. out[tok][:] = is_new ? combined[rel][:] : emb[id][:]
// One 256-thread block per token; b128 loads, non-temporal b128 stores
// (output is written once and never re-read -> keep it out of L2).
// ---------------------------------------------------------------------------
__global__ __launch_bounds__(256) void gather_kernel(
    const int* __restrict__ ids, const int* __restrict__ start_idx,
    const float* __restrict__ emb, const float* __restrict__ combined,
    float* __restrict__ out) {
    const int tok = blockIdx.x;
    const int id  = ids[tok];
    const int start = start_idx[0];

    const v4f* __restrict__ src;
    if (id >= start) {
        int rel = id - start;
        rel = rel < 0 ? 0 : (rel > NUM_NEW - 1 ? NUM_NEW - 1 : rel);
        src = (const v4f*)(combined + (size_t)rel * HIDDEN);
    } else {
        src = (const v4f*)(emb + (size_t)id * HIDDEN);
    }
    v4f* __restrict__ dst = (v4f*)(out + (size_t)tok * HIDDEN);

#pragma unroll
    for (int i = threadIdx.x; i < HIDDEN / 4; i += 256) {
        v4f v = src[i];
        __builtin_nontemporal_store(v, dst + i);
    }
}

// ---------------------------------------------------------------------------
extern "C" void kernel_launch(void* const* d_in, const int* in_sizes, int n_in,
                              void* d_out, int out_size, void* d_ws, size_t ws_size,
                              hipStream_t stream) {
    const int*   ids   = (const int*)  d_in[0];  // [B,S]
    const float* emb   = (const float*)d_in[1];  // [VOCAB+NUM_NEW, HIDDEN]
    const float* sem   = (const float*)d_in[2];  // [NUM_NEW, HIDDEN]
    const float* prof  = (const float*)d_in[3];  // [NUM_NEW, PROF_DIM]
    const float* W1    = (const float*)d_in[4];  // [PROF_DIM, MLP_HID]
    const float* b1    = (const float*)d_in[5];  // [MLP_HID]
    const float* W2    = (const float*)d_in[6];  // [MLP_HID, HIDDEN]
    const float* b2    = (const float*)d_in[7];  // [HIDDEN]
    const int*   start = (const int*)  d_in[8];  // scalar

    float* out      = (float*)d_out;
    float* combined = (float*)d_ws;                              // 512*4096*4 = 8 MB
    float* H        = combined + (size_t)NUM_NEW * HIDDEN;       // + 512*256*4 = 0.5 MB

    // Stage 1: 32 x 16 tiles of [512 x 256]
    mlp1_kernel<<<dim3(NUM_NEW / 16, MLP_HID / 16), 32, 0, stream>>>(prof, W1, b1, H);
    // Stage 2: 32 x 256 tiles of [512 x 4096]
    mlp2_kernel<<<dim3(NUM_NEW / 16, HIDDEN / 16), 32, 0, stream>>>(H, W2, b2, sem, combined);
    // Stage 3: one block per token
    gather_kernel<<<BATCH * SEQ, 256, 0, stream>>>(ids, start, emb, combined, out);
}